// Triaffine_81149112091060
// MI455X (gfx1250) — compile-verified
//
#include <hip/hip_runtime.h>

// ---------------------------------------------------------------------------
// Triaffine factorized contraction for MI455X (gfx1250), fp32 WMMA 16x16x4.
// N=8, L=256, D=256, DP=257.
//   T1[i,b,c] = sum_a l1[i,a] W1[a,b,c]         (stage1 GEMM, per n,i-block)
//   X [j,c]   = h1[i,c] * sum_b l2[j,b] T1[i,b,c]
//   Y [j,c]   = h2[i,c] * sum_b l2[j,b] T2[i,b,c]
//   Z [j,c]   = sum_d Y[j,d] W3[c,d]
//   out[i,j]  = sum_c X[j,c] Z[j,c]             (stages 2+3 fused in LDS)
// ---------------------------------------------------------------------------

#define Dh   256
#define DPd  257
#define NW   65792   // DP * D  (W row length, T row length)
#define KP   260     // DP padded to multiple of 4
#define IBLK 64      // i rows per pass

typedef __attribute__((ext_vector_type(2))) float v2f;
typedef __attribute__((ext_vector_type(8))) float v8f;

__device__ __forceinline__ v8f wmma_f32(v2f a, v2f b, v8f c) {
  // 8 args: (neg_a, A, neg_b, B, c_mod, C, reuse_a, reuse_b)
  return __builtin_amdgcn_wmma_f32_16x16x4_f32(
      false, a, false, b, (short)0, c, false, false);
}

// -------------------------- W3 transpose (256x256) -------------------------
__global__ __launch_bounds__(256) void tri_transpose256(
    const float* __restrict__ W3, float* __restrict__ W3T) {
  __shared__ float t[16][17];
  int bx = blockIdx.x * 16, by = blockIdx.y * 16;
  int x = threadIdx.x, y = threadIdx.y;
  t[y][x] = W3[(by + y) * Dh + bx + x];
  __syncthreads();
  W3T[(bx + y) * Dh + by + x] = t[x][y];  // W3T[d*256+c] = W3[c*256+d]
}

// ------------------- stage 1: T = [l1|1] @ W  (64 x 65792) ------------------
// grid: (65792/64, 2), block: 256 (8 waves). Tile M=32, N=64, K-chunk=52.
#define S1_M   32
#define S1_N   64
#define S1_KCH 52

__global__ __launch_bounds__(256) void tri_stage1(
    const float* __restrict__ Asrc,   // l1 rows for this i-block: [64 x 256]
    const float* __restrict__ W,      // [257 x 65792]
    float* __restrict__ T)            // [64 x 65792]
{
  __shared__ float As[S1_M][S1_KCH];
  __shared__ float Bs[S1_KCH][S1_N];

  const int tid  = threadIdx.x;
  const int wave = tid >> 5;
  const int lane = tid & 31;
  const int half = lane >> 4;
  const int lr   = lane & 15;

  const int n_base = blockIdx.x * S1_N;
  const int m_base = blockIdx.y * S1_M;
  const int mt = wave >> 2;   // 0..1
  const int nt = wave & 3;    // 0..3

  v8f acc = {};

  for (int kb = 0; kb < KP; kb += S1_KCH) {
    // cooperative load: A chunk [32 x 52] with ones column + zero pad
    for (int idx = tid; idx < S1_M * S1_KCH; idx += 256) {
      int m = idx / S1_KCH, kl = idx % S1_KCH, a = kb + kl;
      float v = 0.0f;
      if (a < Dh)        v = Asrc[(m_base + m) * Dh + a];
      else if (a == Dh)  v = 1.0f;
      As[m][kl] = v;
    }
    // cooperative load: B chunk [52 x 64], zero pad past a=256
    for (int idx = tid; idx < S1_KCH * S1_N; idx += 256) {
      int kl = idx / S1_N, nn = idx % S1_N, a = kb + kl;
      Bs[kl][nn] = (a < DPd) ? W[(size_t)a * NW + n_base + nn] : 0.0f;
    }
    __syncthreads();

    // prefetch next K-chunk of the W panel into cache
    int kbn = kb + S1_KCH;
    if (kbn < KP && tid < S1_KCH) {
      int a = kbn + tid;
      if (a < DPd) __builtin_prefetch(&W[(size_t)a * NW + n_base], 0, 1);
    }

#pragma unroll
    for (int k = 0; k < S1_KCH; k += 4) {
      v2f a, b;
      a.x = As[mt * 16 + lr][k + 2 * half + 0];
      a.y = As[mt * 16 + lr][k + 2 * half + 1];
      b.x = Bs[k + 2 * half + 0][nt * 16 + lr];
      b.y = Bs[k + 2 * half + 1][nt * 16 + lr];
      acc = wmma_f32(a, b, acc);
    }
    __syncthreads();
  }

  const int col = n_base + nt * 16 + lr;
#pragma unroll
  for (int r = 0; r < 8; ++r) {
    int m = m_base + mt * 16 + r + 8 * half;
    T[(size_t)m * NW + col] = acc[r];
  }
}

// --------------- stages 2+3 fused: out[i, j0..j0+15] per block --------------
// grid: (16 j-blocks, 64 i), block 256 (8 waves). LDS ~49.6 KB.
#define JT 16

__global__ __launch_bounds__(256) void tri_mega(
    const float* __restrict__ L2n,    // l2 for batch n: [256 x 256]
    const float* __restrict__ T1,     // [64 x 65792]
    const float* __restrict__ T2,     // [64 x 65792]
    const float* __restrict__ h1n,    // [256 x 256]
    const float* __restrict__ h2n,    // [256 x 256]
    const float* __restrict__ W3T,    // [256 x 256]  (d-major)
    float* __restrict__ outn,         // [256 x 256]  rows i
    int i0)
{
  __shared__ float Al[JT][KP];        // l2 tile (+ones,+pad): 16.6 KB
  __shared__ float Xs[JT][Dh + 1];    // 16.1 KB (padded stride vs banks)
  __shared__ float Ys[JT][Dh + 1];    // 16.1 KB
  __shared__ float wpart[8][JT];      // per-wave out partials (deterministic)

  const int tid  = threadIdx.x;
  const int wave = tid >> 5;
  const int lane = tid & 31;
  const int half = lane >> 4;
  const int lr   = lane & 15;

  const int j0 = blockIdx.x * JT;
  const int il = blockIdx.y;          // 0..63
  const int i  = i0 + il;

  const float* T1r = T1 + (size_t)il * NW;
  const float* T2r = T2 + (size_t)il * NW;

  // load l2 tile [16 x 260] with ones column + zero pad
  for (int idx = tid; idx < JT * KP; idx += 256) {
    int j = idx / KP, b = idx % KP;
    float v = 0.0f;
    if (b < Dh)        v = L2n[(j0 + j) * Dh + b];
    else if (b == Dh)  v = 1.0f;
    Al[j][b] = v;
  }
  __syncthreads();

  // ---- phase A: X = h1 .* (l2 @ T1),  Y = h2 .* (l2 @ T2) ----
  for (int side = 0; side < 2; ++side) {
    const float* Trow = side ? T2r : T1r;
    const float* hb   = side ? h2n : h1n;
    float (*S)[Dh + 1] = side ? Ys : Xs;

#pragma unroll
    for (int t = 0; t < 2; ++t) {
      const int c0 = (wave + t * 8) * 16;
      const int cc = c0 + lr;
      v8f acc = {};
      for (int kk = 0; kk < KP; kk += 4) {
        v2f a, b;
        a.x = Al[lr][kk + 2 * half + 0];
        a.y = Al[lr][kk + 2 * half + 1];
        int b0 = kk + 2 * half;
        b.x = (b0 < DPd)     ? Trow[b0 * Dh + cc]       : 0.0f;
        b.y = (b0 + 1 < DPd) ? Trow[(b0 + 1) * Dh + cc] : 0.0f;
        acc = wmma_f32(a, b, acc);
      }
      const float s = hb[i * Dh + cc];
#pragma unroll
      for (int r = 0; r < 8; ++r)
        S[r + 8 * half][cc] = acc[r] * s;
    }
    __syncthreads();
  }

  // ---- phase B: Z = Y @ W3T, out[j] = sum_c X[j,c]*Z[j,c] ----
  float prods[8] = {0.f, 0.f, 0.f, 0.f, 0.f, 0.f, 0.f, 0.f};
#pragma unroll
  for (int t = 0; t < 2; ++t) {
    const int c0 = (wave + t * 8) * 16;
    const int cc = c0 + lr;
    v8f acc = {};
    for (int kk = 0; kk < Dh; kk += 4) {
      v2f a, b;
      a.x = Ys[lr][kk + 2 * half + 0];
      a.y = Ys[lr][kk + 2 * half + 1];
      int d0 = kk + 2 * half;
      b.x = W3T[d0 * Dh + cc];
      b.y = W3T[(d0 + 1) * Dh + cc];
      acc = wmma_f32(a, b, acc);
    }
#pragma unroll
    for (int r = 0; r < 8; ++r)
      prods[r] += acc[r] * Xs[r + 8 * half][cc];
  }

  // deterministic reduction: shuffle within 16-lane halves, then per-wave LDS
#pragma unroll
  for (int r = 0; r < 8; ++r) {
    float p = prods[r];
    p += __shfl_xor(p, 8, 16);
    p += __shfl_xor(p, 4, 16);
    p += __shfl_xor(p, 2, 16);
    p += __shfl_xor(p, 1, 16);
    if (lr == 0) wpart[wave][r + 8 * half] = p;
  }
  __syncthreads();

  if (tid < JT) {
    float s = 0.0f;
#pragma unroll
    for (int w = 0; w < 8; ++w) s += wpart[w][tid];
    outn[i * Dh + j0 + tid] = s;
  }
}

// ---------------------------------------------------------------------------
extern "C" void kernel_launch(void* const* d_in, const int* in_sizes, int n_in,
                              void* d_out, int out_size, void* d_ws, size_t ws_size,
                              hipStream_t stream) {
  (void)in_sizes; (void)n_in; (void)out_size; (void)ws_size;
  const float* layer1 = (const float*)d_in[0];
  const float* layer2 = (const float*)d_in[1];
  const float* h1     = (const float*)d_in[2];
  const float* h2     = (const float*)d_in[3];
  const float* W1     = (const float*)d_in[4];
  const float* W2     = (const float*)d_in[5];
  const float* W3     = (const float*)d_in[6];
  float* out = (float*)d_out;

  // workspace layout: W3T (64K floats) | T1 (64*65792) | T2 (64*65792) ~ 34 MB
  float* W3T = (float*)d_ws;
  float* T1  = W3T + 65536;
  float* T2  = T1 + (size_t)IBLK * NW;

  tri_transpose256<<<dim3(16, 16), dim3(16, 16), 0, stream>>>(W3, W3T);

  for (int n = 0; n < 8; ++n) {
    for (int ib = 0; ib < 4; ++ib) {
      const int i0 = ib * IBLK;
      const float* Ablk = layer1 + ((size_t)n * 256 + i0) * Dh;
      tri_stage1<<<dim3(NW / S1_N, 2), 256, 0, stream>>>(Ablk, W1, T1);
      tri_stage1<<<dim3(NW / S1_N, 2), 256, 0, stream>>>(Ablk, W2, T2);
      tri_mega<<<dim3(16, IBLK), 256, 0, stream>>>(
          layer2 + (size_t)n * 256 * Dh, T1, T2,
          h1 + (size_t)n * 256 * Dh, h2 + (size_t)n * 256 * Dh,
          W3T, out + (size_t)n * 256 * 256, i0);
    }
  }
}